// FractalConsciousnessProcessor_34213709480831
// MI455X (gfx1250) — compile-verified
//
#include <hip/hip_runtime.h>

#define BATCH 2048
#define SEQ   64
#define EMBED 512
#define NK    20
#define DT    0.01f
#define INV2PI 0.15915494309189535f

typedef float v2f __attribute__((ext_vector_type(2)));
typedef float v8f __attribute__((ext_vector_type(8)));

__device__ __forceinline__ float wave_sum(float v) {
#pragma unroll
  for (int o = 16; o > 0; o >>= 1) v += __shfl_xor(v, o, 32);
  return v;
}
__device__ __forceinline__ float wave_max(float v) {
#pragma unroll
  for (int o = 16; o > 0; o >>= 1) v = fmaxf(v, __shfl_xor(v, o, 32));
  return v;
}

// Block-wide reduction over 16 waves. Exactly one barrier; caller cycles
// through 3 scratch arrays so a read of array X is always separated from the
// next write of X by >= 1 later barrier.
__device__ __forceinline__ float block_sum(float v, float* red, int lane, int wave) {
  v = wave_sum(v);
  if (lane == 0) red[wave] = v;
  __syncthreads();
  float t = red[0];
#pragma unroll
  for (int i = 1; i < 16; ++i) t += red[i];
  return t;
}
__device__ __forceinline__ float block_max(float v, float* red, int lane, int wave) {
  v = wave_max(v);
  if (lane == 0) red[wave] = v;
  __syncthreads();
  float t = red[0];
#pragma unroll
  for (int i = 1; i < 16; ++i) t = fmaxf(t, red[i]);
  return t;
}

// F(psi) = sum_k lam_k * sin(2*pi*(k+1)*psi + omega_k*t + PHASE)
// computed in "turns" so v_sin_f32 needs no scaling: sin(2*pi*turn).
__device__ __forceinline__ float field_at(float psi, float t, const float* lam_r) {
  float f = 0.f;
#pragma unroll
  for (int k = 0; k < NK; ++k) {
    const float omega_t = (0.5f + (float)k * (4.5f / 19.f)) * INV2PI;  // omega_k / 2pi
    const float phase_t = 0.7854f * INV2PI;                            // PHASE / 2pi
    float ck   = fmaf(t, omega_t, phase_t);
    float turn = fmaf((float)(k + 1), psi, ck);
    f = fmaf(lam_r[k], __builtin_amdgcn_sinf(turn), f);
  }
  return f;
}

__global__ void __launch_bounds__(512)
fcp_main(const float* __restrict__ x, const float* __restrict__ lam,
         const float* __restrict__ chaos_u,
         float* __restrict__ psi_out, float* __restrict__ field_out,
         float* __restrict__ traj, float* __restrict__ D_out,
         float* __restrict__ ent_ws, int num_steps) {
  __shared__ float sh_psi[EMBED];
  __shared__ float sh_fp[EMBED];
  __shared__ float redA[16], redB[16], redC[16];

  const int b    = blockIdx.x;
  const int tid  = threadIdx.x;      // 512 threads = 16 wave32
  const int lane = tid & 31;
  const int wave = tid >> 5;
  const int d    = tid;

  float lam_r[NK];
#pragma unroll
  for (int k = 0; k < NK; ++k) lam_r[k] = lam[k];   // uniform -> scalar loads

  // ---- init: S_d = sum_s x[b,s,d] via chained V_WMMA_F32_16X16X4_F32 ----
  // A = ones(16x4)  =>  D[m,n] = sum_k B[k,n]  (invariant to K-slot layout).
  const float* xb = x + (size_t)b * (SEQ * EMBED);
  v2f aOne; aOne[0] = 1.f; aOne[1] = 1.f;          // A[m][k] = 1 for all slots
  v8f acc0 = {0.f,0.f,0.f,0.f,0.f,0.f,0.f,0.f};
  v8f acc1 = acc0;
  const int half2 = (lane >> 4) * 2;               // K half owned by this lane
  const int n16   = lane & 15;                     // column within tile
  const int col0  = wave * 32 + n16;               // tile0 column; tile1 = +16
#pragma unroll
  for (int s0 = 0; s0 < SEQ; s0 += 4) {
    v2f b0, b1;
    b0[0] = xb[(s0 + half2 + 0) * EMBED + col0];
    b0[1] = xb[(s0 + half2 + 1) * EMBED + col0];
    b1[0] = xb[(s0 + half2 + 0) * EMBED + col0 + 16];
    b1[1] = xb[(s0 + half2 + 1) * EMBED + col0 + 16];
    acc0 = __builtin_amdgcn_wmma_f32_16x16x4_f32(false, aOne, false, b0,
                                                 (short)0, acc0, false, false);
    acc1 = __builtin_amdgcn_wmma_f32_16x16x4_f32(false, aOne, false, b1,
                                                 (short)0, acc1, false, false);
  }
  // C/D VGPR0: lanes 0-15 = row M=0 of tile, lanes 16-31 = row M=8 (all rows
  // equal to the column sum), so each lane's own d = wave*32+lane is covered.
  float S = (lane < 16) ? acc0[0] : acc1[0];

  // softmax over embed
  float mx = block_max(S, redA, lane, wave);
  float e  = __expf(S - mx);
  float se = block_sum(e, redB, lane, wave);

  // chaotic logistic-map modulation
  float z = fmaf(chaos_u[(size_t)b * EMBED + d], 0.5f, 0.25f);
#pragma unroll
  for (int i = 0; i < 10; ++i) z = 3.9f * z * (1.f - z);
  float mod = fmaf(0.5f, __builtin_amdgcn_sinf(z), 0.5f);  // 0.5 + 0.5*sin(2pi z)

  float p   = (e / se) * mod;
  float sp  = block_sum(p, redC, lane, wave);
  float psi = p * __builtin_amdgcn_rcpf(sp);

  const int dm = (d == 0) ? (EMBED - 1) : d - 1;
  const int dp = (d == EMBED - 1) ? 0 : d + 1;

  for (int step = 0; step < num_steps; ++step) {
    float t     = (float)step * DT;
    float field = field_at(psi, t, lam_r);
    float fp    = field * psi;
    sh_psi[d] = psi;
    sh_fp[d]  = fp;
    float R1 = block_sum(fp, redA, lane, wave);   // barrier also publishes sh_*
    float coupling = R1 * (1.f / (float)EMBED);
    float Dc = fmaf(9.99f, __builtin_amdgcn_rcpf(1.f + __expf(-coupling)), 0.01f);
    float divv = (sh_fp[dp] - sh_fp[dm]) * 0.5f;
    float lap  = sh_psi[dp] + sh_psi[dm] - 2.f * psi;
    float nv   = fmaf(DT, fmaf(Dc, lap, -divv), psi);
    nv = fmaxf(nv, 1e-10f);
    float R2 = block_sum(nv, redB, lane, wave);
    float pn = nv * __builtin_amdgcn_rcpf(R2);
    __builtin_nontemporal_store(pn, &traj[((size_t)step * BATCH + b) * EMBED + d]);
    float ent = -pn * __logf(pn + 1e-10f);
    float R3 = block_sum(ent, redC, lane, wave);
    if (tid == 0) ent_ws[(size_t)step * BATCH + b] = R3;
    psi = pn;
  }

  // final field / diffusion at t = num_steps*DT
  float tF     = (float)num_steps * DT;
  float fieldF = field_at(psi, tF, lam_r);
  float RF = block_sum(fieldF * psi, redA, lane, wave);
  float DcF = fmaf(9.99f,
      __builtin_amdgcn_rcpf(1.f + __expf(-RF * (1.f / (float)EMBED))), 0.01f);
  psi_out[(size_t)b * EMBED + d]   = psi;
  field_out[(size_t)b * EMBED + d] = fieldF;
  if (tid == 0) D_out[b] = DcF;
}

// fci[s] = mean_b(ent_ws[s,b]) / ln(512)
__global__ void __launch_bounds__(256)
fcp_fci(const float* __restrict__ ent_ws, float* __restrict__ fci) {
  __shared__ float red[8];
  const int s = blockIdx.x;
  const int tid = threadIdx.x;
  float a = 0.f;
  for (int i = tid; i < BATCH; i += 256) a += ent_ws[(size_t)s * BATCH + i];
  a = wave_sum(a);
  if ((tid & 31) == 0) red[tid >> 5] = a;
  __syncthreads();
  if (tid == 0) {
    float t = red[0];
#pragma unroll
    for (int i = 1; i < 8; ++i) t += red[i];
    fci[s] = t * (1.f / (float)BATCH) * (1.f / 6.2383246250395077f); // /ln(512)
  }
}

extern "C" void kernel_launch(void* const* d_in, const int* in_sizes, int n_in,
                              void* d_out, int out_size, void* d_ws, size_t ws_size,
                              hipStream_t stream) {
  const float* x   = (const float*)d_in[0];
  const float* lam = (const float*)d_in[1];
  const float* u   = (const float*)d_in[2];
  const long long BD = (long long)BATCH * EMBED;
  // out = psi_f[BD] | field[BD] | traj[ns*BD] | fci[ns] | D[BATCH]
  int ns = (int)(((long long)out_size - 2 * BD - BATCH) / (BD + 1));
  float* out   = (float*)d_out;
  float* psi_f = out;
  float* field = out + BD;
  float* traj  = out + 2 * BD;
  float* fci   = traj + (long long)ns * BD;
  float* Dv    = fci + ns;
  float* ws    = (float*)d_ws;  // ns*BATCH floats of per-row entropy

  fcp_main<<<BATCH, 512, 0, stream>>>(x, lam, u, psi_f, field, traj, Dv, ws, ns);
  fcp_fci<<<ns, 256, 0, stream>>>(ws, fci);
}